// Decoder_18425409700535
// MI455X (gfx1250) — compile-verified
//
#include <hip/hip_runtime.h>
#include <stdint.h>

// Sequential LSTM scan (P=81 steps).
//   Zu = h_enc @ U + b precomputed once (removes U from the serial chain).
//   Per step p: z = h @ W + Zu[p]; i,f,o=sigmoid, g=identity; c=f*h+i*g; h'=o*c;
//   orders[p] = pre-update h.
// fp32 matmul via V_WMMA_F32_16X16X4_F32 (wave32). W+U (134MB) stay L2-resident
// (192MB global L2), so per-step W traffic is L2 hits. A-panel (h state) is
// staged into LDS with double-buffered GLOBAL_LOAD_ASYNC_TO_LDS_B128 (ASYNCcnt)
// so staging overlaps WMMA compute.

#define P_PLACES 81
#define H_DIM    2048
#define H4_DIM   8192
#define MROWS    48           // rows per WG (96 padded rows split across blockIdx.y)
#define MT_W     3            // 3 M-tiles of 16 per wave
#define KC       128          // K chunk staged in LDS
#define LSTR     132          // LDS row stride (floats): 528B rows, 16B aligned, bank-skewed
#define TPB      128          // 4 waves = 4 gates (or 4 col slots)
#define NCHUNK   (H_DIM / KC) // 16
#define QPT      ((MROWS * (KC / 4)) / TPB)  // float4 copies per thread per chunk = 12

typedef __attribute__((ext_vector_type(2))) float v2f;
typedef __attribute__((ext_vector_type(8))) float v8f;

__device__ __forceinline__ uint32_t lds_offset(const void* p) {
  // Low 32 bits of a generic pointer into LDS == byte offset from LDS base.
  return (uint32_t)(uintptr_t)p;
}

__device__ __forceinline__ void async_cp_b128(uint32_t ldsoff, const float* g) {
  // CDNA5 async direct-to-LDS copy, tracked by ASYNCcnt (16B per lane).
  asm volatile("global_load_async_to_lds_b128 %0, %1, off"
               :: "v"(ldsoff), "v"(g) : "memory");
}

__device__ __forceinline__ void wait_async_le12() {
  asm volatile("s_wait_asynccnt 0xc" ::: "memory");
}
__device__ __forceinline__ void wait_async_0() {
  asm volatile("s_wait_asynccnt 0x0" ::: "memory");
}

// Issue async copies for A[rowbase : rowbase+48, kc : kc+KC] -> buf (padded LSTR).
// Rows >= 81 are clamped to row 80: keeps per-wave ASYNCcnt at exactly QPT and
// avoids OOB; the duplicated data lands in padded M rows whose results are never
// stored.
__device__ __forceinline__ void issue_chunk(const float* __restrict__ A, int rowbase,
                                            int kc, float* buf) {
  const uint32_t base = lds_offset(buf);
#pragma unroll
  for (int i = 0; i < QPT; ++i) {
    const int idx  = threadIdx.x + i * TPB;   // quad index within 48x32 quads
    const int lrow = idx >> 5;                // 32 float4 per row
    const int q    = idx & 31;
    int grow = rowbase + lrow;
    grow = (grow > P_PLACES - 1) ? (P_PLACES - 1) : grow;
    const float* g = A + (size_t)grow * H_DIM + kc + q * 4;
    async_cp_b128(base + (uint32_t)(lrow * LSTR + q * 4) * 4u, g);
  }
}

// acc[t] += A[rowbase + t*16 .. +16 rows, K=0..2047] x B[:, colbase..colbase+15]
// A: [<=81, 2048] row-major (logical rows >=81 are don't-care). B: [2048, 8192].
__device__ __forceinline__ void gemm_panel(const float* __restrict__ A,
                                           const float* __restrict__ B,
                                           int colbase, int rowbase,
                                           float* lds,          // 2 * MROWS * LSTR floats
                                           v8f acc[MT_W]) {
  const int lane = threadIdx.x & 31;
  const int half = lane >> 4;   // 0: K,K+1 ; 1: K+2,K+3
  const int l16  = lane & 15;
  const v8f vzero = {0.f, 0.f, 0.f, 0.f, 0.f, 0.f, 0.f, 0.f};
#pragma unroll
  for (int t = 0; t < MT_W; ++t) acc[t] = vzero;

  float* buf0 = lds;
  float* buf1 = lds + MROWS * LSTR;

  issue_chunk(A, rowbase, 0, buf0);
  for (int c = 0; c < NCHUNK; ++c) {
    if (c + 1 < NCHUNK) {
      issue_chunk(A, rowbase, (c + 1) * KC, ((c + 1) & 1) ? buf1 : buf0);
      wait_async_le12();   // <=12 outstanding => chunk c complete (in-order)
    } else {
      wait_async_0();
    }
    __syncthreads();       // all waves' chunk-c data visible

    const float* cur = (c & 1) ? buf1 : buf0;
    const int kcb = c * KC;
#pragma unroll 4
    for (int kk = 0; kk < KC; kk += 4) {
      // B fragment (4x16): lanes 0-15 rows k,k+1 ; lanes 16-31 rows k+2,k+3
      const int br = kcb + kk + half * 2;
      v2f bf;
      bf.x = B[(size_t)br * H4_DIM + colbase + l16];
      bf.y = B[(size_t)(br + 1) * H4_DIM + colbase + l16];
#pragma unroll
      for (int t = 0; t < MT_W; ++t) {
        const float* ap = cur + (t * 16 + l16) * LSTR + kk + half * 2;
        v2f af = *(const v2f*)ap;     // ds_load_b64, rows skewed 4 banks apart
        acc[t] = __builtin_amdgcn_wmma_f32_16x16x4_f32(
            false, af, false, bf, (short)0, acc[t], false, false);
      }
    }
    __syncthreads();       // all waves done reading `cur` before it is re-filled
  }
}

// Zu[p, j] = sum_k h_enc[p,k] * U[k,j] + b[j]   -> [81, 8192]
__global__ void __launch_bounds__(TPB) zu_kernel(const float* __restrict__ h_enc,
                                                 const float* __restrict__ U,
                                                 const float* __restrict__ b,
                                                 float* __restrict__ Zu) {
  __shared__ float lds[2 * MROWS * LSTR];
  const int wave = threadIdx.x >> 5;
  const int lane = threadIdx.x & 31;
  const int colbase = blockIdx.x * 64 + wave * 16;   // grid.x = 8192/64 = 128
  const int rowbase = blockIdx.y * MROWS;            // grid.y = 2

  v8f acc[MT_W];
  gemm_panel(h_enc, U, colbase, rowbase, lds, acc);

  const int col = colbase + (lane & 15);
  const int rb  = (lane >> 4) * 8;
  const float bc = b[col];
#pragma unroll
  for (int t = 0; t < MT_W; ++t)
#pragma unroll
    for (int r = 0; r < 8; ++r) {
      const int row = rowbase + t * 16 + rb + r;
      if (row < P_PLACES) Zu[(size_t)row * H4_DIM + col] = acc[t][r] + bc;
    }
}

__device__ __forceinline__ float sigm(float x) { return 1.f / (1.f + __expf(-x)); }

// One scan step: z = h_in @ W + Zu_p ; h_out = sig(zo)*(sig(zf)*h_in + sig(zi)*zg);
// orders_p = h_in (pre-update state).
__global__ void __launch_bounds__(TPB) lstm_step_kernel(const float* __restrict__ h_in,
                                                        const float* __restrict__ W,
                                                        const float* __restrict__ Zu_p,
                                                        float* __restrict__ h_out,
                                                        float* __restrict__ orders_p) {
  __shared__ float lds[2 * MROWS * LSTR];  // reused as zbuf[4][48][16] afterwards
  const int wave = threadIdx.x >> 5;       // gate: 0=i 1=f 2=g 3=o
  const int lane = threadIdx.x & 31;
  const int n0 = blockIdx.x * 16;          // grid.x = 2048/16 = 128
  const int rowbase = blockIdx.y * MROWS;  // grid.y = 2

  v8f acc[MT_W];
  gemm_panel(h_in, W, wave * H_DIM + n0, rowbase, lds, acc);
  // gemm_panel ends with __syncthreads(): safe to repurpose lds.

  float* zbuf = lds;                       // [4][MROWS][16] = 12KB
  {
    const int col = lane & 15;
    const int rb  = (lane >> 4) * 8;
#pragma unroll
    for (int t = 0; t < MT_W; ++t)
#pragma unroll
      for (int r = 0; r < 8; ++r)
        zbuf[(wave * MROWS + t * 16 + rb + r) * 16 + col] = acc[t][r];
  }
  __syncthreads();

  for (int e = threadIdx.x; e < MROWS * 16; e += TPB) {
    const int lm = e >> 4;
    const int c  = e & 15;
    const int m  = rowbase + lm;
    if (m < P_PLACES) {
      const float zi = zbuf[(0 * MROWS + lm) * 16 + c] + Zu_p[0 * H_DIM + n0 + c];
      const float zf = zbuf[(1 * MROWS + lm) * 16 + c] + Zu_p[1 * H_DIM + n0 + c];
      const float zg = zbuf[(2 * MROWS + lm) * 16 + c] + Zu_p[2 * H_DIM + n0 + c];
      const float zo = zbuf[(3 * MROWS + lm) * 16 + c] + Zu_p[3 * H_DIM + n0 + c];
      const float hp = h_in[(size_t)m * H_DIM + n0 + c];
      const float cn = sigm(zf) * hp + sigm(zi) * zg;  // identity g
      const float hn = sigm(zo) * cn;                  // identity output act
      orders_p[(size_t)m * H_DIM + n0 + c] = hp;
      h_out  [(size_t)m * H_DIM + n0 + c] = hn;
    }
  }
}

extern "C" void kernel_launch(void* const* d_in, const int* in_sizes, int n_in,
                              void* d_out, int out_size, void* d_ws, size_t ws_size,
                              hipStream_t stream) {
  const float* h_enc      = (const float*)d_in[0];
  // d_in[1] = mask (unused, as in reference)
  const float* h_dec_init = (const float*)d_in[2];
  const float* W          = (const float*)d_in[3];
  const float* U          = (const float*)d_in[4];
  const float* b          = (const float*)d_in[5];
  float* out = (float*)d_out;

  float* Zu = (float*)d_ws;                        // 81*8192 f32
  float* hA = Zu + (size_t)P_PLACES * H4_DIM;      // 81*2048 f32
  float* hB = hA + (size_t)P_PLACES * H_DIM;       // 81*2048 f32 (~4 MB total ws)

  zu_kernel<<<dim3(H4_DIM / 64, 2), TPB, 0, stream>>>(h_enc, U, b, Zu);

  const float* hin = h_dec_init;
  float* hout = hA;
  for (int p = 0; p < P_PLACES; ++p) {
    lstm_step_kernel<<<dim3(H_DIM / 16, 2), TPB, 0, stream>>>(
        hin, W, Zu + (size_t)p * H4_DIM, hout,
        out + (size_t)p * (size_t)P_PLACES * H_DIM);
    hin  = hout;
    hout = (hout == hA) ? hB : hA;
  }
}